// Extractor_56564719288936
// MI455X (gfx1250) — compile-verified
//
#include <hip/hip_runtime.h>

typedef float v2f __attribute__((ext_vector_type(2)));
typedef float v4f __attribute__((ext_vector_type(4)));
typedef float v8f __attribute__((ext_vector_type(8)));

#define HW    76800      // 240*320 pixels (b=1)
#define W_IMG 320
#define VX    320
#define VY    320
#define VZ    320
#define NPTS  9

// Output layout (floats), concatenated in reference return order:
//   fv     : [0,            9*HW)
//   fw     : [9*HW,        18*HW)
//   ray_pts: [18*HW,       45*HW)   (9*3 per pixel)
//   depth  : [45*HW,       46*HW)
//   inds   : [46*HW,      262*HW)   (9*8*3 per pixel, integer values stored as f32)
//   w8     : [262*HW,     334*HW)   (9*8 per pixel)
//   coords : [334*HW,     337*HW)
__global__ __launch_bounds__(256) void Extractor_56564719288936_kernel(
    const float* __restrict__ depth,
    const float* __restrict__ extr,     // 4x4
    const float* __restrict__ intr,     // 3x3
    const float* __restrict__ volume,   // 320^3
    const float* __restrict__ wvol,     // 320^3
    const float* __restrict__ origin,   // 3
    const float* __restrict__ resptr,   // 1
    float* __restrict__ out)
{
    const int  p    = blockIdx.x * 256 + (int)threadIdx.x;   // pixel id (exact grid)
    const int  lane = (int)threadIdx.x & 31;
    const bool lo   = lane < 16;
    const int  m    = lane & 15;

    // ---------- uniform small matrices ----------
    const float K00=intr[0],K01=intr[1],K02=intr[2],
                K10=intr[3],K11=intr[4],K12=intr[5],
                K20=intr[6],K21=intr[7],K22=intr[8];
    const float det = K00*(K11*K22-K12*K21) - K01*(K10*K22-K12*K20) + K02*(K10*K21-K11*K20);
    float iK[3][3];
    iK[0][0]= (K11*K22-K12*K21)/det; iK[0][1]=-(K01*K22-K02*K21)/det; iK[0][2]= (K01*K12-K02*K11)/det;
    iK[1][0]=-(K10*K22-K12*K20)/det; iK[1][1]= (K00*K22-K02*K20)/det; iK[1][2]=-(K00*K12-K02*K10)/det;
    iK[2][0]= (K10*K21-K11*K20)/det; iK[2][1]=-(K00*K21-K01*K20)/det; iK[2][2]= (K00*K11-K01*K10)/det;

    float E[4][4];
    #pragma unroll
    for (int r=0;r<4;++r)
      #pragma unroll
      for (int c=0;c<4;++c) E[r][c]=extr[r*4+c];

    // T = E * [[invK,0],[0,1]]  (4x4); world = T * (u*z, v*z, z, 1)
    float T[4][4];
    #pragma unroll
    for (int r=0;r<4;++r){
      #pragma unroll
      for (int c=0;c<3;++c)
        T[r][c] = E[r][0]*iK[0][c] + E[r][1]*iK[1][c] + E[r][2]*iK[2][c];
      T[r][3] = E[r][3];
    }

    // ---------- per-pixel homogeneous pinhole point ----------
    const float z   = depth[p];
    const float uco = (float)(p % W_IMG);   // col (yy in reference)
    const float vro = (float)(p / W_IMG);   // row (xx in reference)
    const float pxh = uco * z;              // u*z
    const float pyh = vro * z;              // v*z

    // ---------- A operand: T in rows 0..3 of a 16x4 f32 matrix ----------
    // ISA layout: v0 = K0 (lanes 0-15) / K2 (lanes 16-31); v1 = K1 / K3; M = lane%16.
    v2f a;
    a[0] = (m==0) ? (lo?T[0][0]:T[0][2])
         : (m==1) ? (lo?T[1][0]:T[1][2])
         : (m==2) ? (lo?T[2][0]:T[2][2])
         : (m==3) ? (lo?T[3][0]:T[3][2]) : 0.0f;
    a[1] = (m==0) ? (lo?T[0][1]:T[0][3])
         : (m==1) ? (lo?T[1][1]:T[1][3])
         : (m==2) ? (lo?T[2][1]:T[2][3])
         : (m==3) ? (lo?T[3][1]:T[3][3]) : 0.0f;

    v8f czero = {};

    // ---------- B operand (4x16): columns = 16 pixels' (u*z, v*z, z, 1) ----------
    // Assumed B layout mirrors A/C: v0 = row K0 (lanes 0-15) / K2 (lanes 16-31); v1 = K1/K3.
    // Group 0: pixels in lanes 0..15 of this wave.
    float sx0=__shfl(pxh,m), sy0=__shfl(pyh,m), sz0=__shfl(z,m);
    v2f b0; b0[0] = lo ? sx0 : sz0; b0[1] = lo ? sy0 : 1.0f;
    v8f d0 = __builtin_amdgcn_wmma_f32_16x16x4_f32(false, a, false, b0, (short)0, czero, false, false);

    // Group 1: pixels in lanes 16..31.
    float sx1=__shfl(pxh,m+16), sy1=__shfl(pyh,m+16), sz1=__shfl(z,m+16);
    v2f b1; b1[0] = lo ? sx1 : sz1; b1[1] = lo ? sy1 : 1.0f;
    v8f d1 = __builtin_amdgcn_wmma_f32_16x16x4_f32(false, a, false, b1, (short)0, czero, false, false);

    // D rows 0..2 (VGPRs 0..2, lanes 0-15 hold columns 0-15) -> this pixel's world coords.
    float g0=__shfl(d1[0],m), g1=__shfl(d1[1],m), g2=__shfl(d1[2],m);
    const float cwx = lo ? d0[0] : g0;
    const float cwy = lo ? d0[1] : g1;
    const float cwz = lo ? d0[2] : g2;

    // ---------- voxel-space center / eye / unit direction ----------
    const float ox=origin[0], oy=origin[1], oz=origin[2];
    const float res = resptr[0];
    const float cx = (cwx - ox) / res;
    const float cy = (cwy - oy) / res;
    const float cz = (cwz - oz) / res;
    const float evx = (E[0][3] - ox) / res;
    const float evy = (E[1][3] - oy) / res;
    const float evz = (E[2][3] - oz) / res;
    float dx = cx - evx, dy = cy - evy, dz = cz - evz;
    const float nrm = sqrtf(dx*dx + dy*dy + dz*dz);
    const float dn  = fmaxf(nrm, 1e-12f);
    dx /= dn; dy /= dn; dz /= dn;

    // ---------- output regions ----------
    float* __restrict__ fvO = out;
    float* __restrict__ fwO = out + (size_t)9*HW;
    float* __restrict__ rpO = out + (size_t)18*HW;
    float* __restrict__ dpO = out + (size_t)45*HW;
    float* __restrict__ inO = out + (size_t)46*HW;
    float* __restrict__ w8O = out + (size_t)262*HW;
    float* __restrict__ cdO = out + (size_t)334*HW;

    __builtin_nontemporal_store(cwx, cdO + (size_t)p*3 + 0);
    __builtin_nontemporal_store(cwy, cdO + (size_t)p*3 + 1);
    __builtin_nontemporal_store(cwz, cdO + (size_t)p*3 + 2);
    __builtin_nontemporal_store(z,   dpO + p);

    for (int t = 0; t < NPTS; ++t) {
        const float off = (float)(t - 4);
        const float ptx = cx + off*dx;
        const float pty = cy + off*dy;
        const float ptz = cz + off*dz;

        const size_t pt = (size_t)p*NPTS + t;
        __builtin_nontemporal_store(ptx, rpO + pt*3 + 0);
        __builtin_nontemporal_store(pty, rpO + pt*3 + 1);
        __builtin_nontemporal_store(ptz, rpO + pt*3 + 2);

        // per-dimension floor / neighbor-sign / alpha
        const float flx = floorf(ptx), fly = floorf(pty), flz = floorf(ptz);
        const float dfx = (flx+0.5f)-ptx, dfy = (fly+0.5f)-pty, dfz = (flz+0.5f)-ptz;
        const float nbx = (dfx>0.f)?1.f:((dfx<0.f)?-1.f:0.f);
        const float nby = (dfy>0.f)?1.f:((dfy<0.f)?-1.f:0.f);
        const float nbz = (dfz>0.f)?1.f:((dfz<0.f)?-1.f:0.f);
        const float alx = fabsf(ptx-(flx+0.5f));
        const float aly = fabsf(pty-(fly+0.5f));
        const float alz = fabsf(ptz-(flz+0.5f));

        float fvacc = 0.f, fwacc = 0.f;
        float indsF[24], w8F[8];

        #pragma unroll
        for (int c = 0; c < 8; ++c) {
            const int bi = (c>>2)&1, bj = (c>>1)&1, bk = c&1;
            const float wx8 = bi ? alx : 1.f-alx;
            const float wy8 = bj ? aly : 1.f-aly;
            const float wz8 = bk ? alz : 1.f-alz;
            const float wcur = (wx8*wy8)*wz8;

            const float ix = bi ? flx+nbx : flx;
            const float iy = bj ? fly+nby : fly;
            const float iz = bk ? flz+nbz : flz;
            const bool valid = (ix>=0.f)&&(ix<(float)VX)&&(iy>=0.f)&&(iy<(float)VY)&&(iz>=0.f)&&(iz<(float)VZ);

            const int iix = (int)fminf(fmaxf(ix,0.f),(float)(VX-1));
            const int iiy = (int)fminf(fmaxf(iy,0.f),(float)(VY-1));
            const int iiz = (int)fminf(fmaxf(iz,0.f),(float)(VZ-1));
            const int lin = (iix*VY + iiy)*VZ + iiz;

            const float vv = volume[lin];
            const float wv = wvol[lin];
            fvacc += (valid ? vv : 0.f) * wcur;
            fwacc += (valid ? wv : 0.f) * wcur;

            indsF[c*3+0] = (float)iix;
            indsF[c*3+1] = (float)iiy;
            indsF[c*3+2] = (float)iiz;
            w8F[c] = wcur;
        }

        __builtin_nontemporal_store(fvacc, fvO + pt);
        __builtin_nontemporal_store(fwacc, fwO + pt);

        v4f* ip = (v4f*)(inO + pt*24);   // 24 floats/(p,t), base 16B-aligned
        v4f q0 = {indsF[ 0],indsF[ 1],indsF[ 2],indsF[ 3]};
        v4f q1 = {indsF[ 4],indsF[ 5],indsF[ 6],indsF[ 7]};
        v4f q2 = {indsF[ 8],indsF[ 9],indsF[10],indsF[11]};
        v4f q3 = {indsF[12],indsF[13],indsF[14],indsF[15]};
        v4f q4 = {indsF[16],indsF[17],indsF[18],indsF[19]};
        v4f q5 = {indsF[20],indsF[21],indsF[22],indsF[23]};
        __builtin_nontemporal_store(q0, ip+0);
        __builtin_nontemporal_store(q1, ip+1);
        __builtin_nontemporal_store(q2, ip+2);
        __builtin_nontemporal_store(q3, ip+3);
        __builtin_nontemporal_store(q4, ip+4);
        __builtin_nontemporal_store(q5, ip+5);

        v4f* wp = (v4f*)(w8O + pt*8);    // 8 floats/(p,t), base 16B-aligned
        v4f w0 = {w8F[0],w8F[1],w8F[2],w8F[3]};
        v4f w1 = {w8F[4],w8F[5],w8F[6],w8F[7]};
        __builtin_nontemporal_store(w0, wp+0);
        __builtin_nontemporal_store(w1, wp+1);
    }
}

extern "C" void kernel_launch(void* const* d_in, const int* in_sizes, int n_in,
                              void* d_out, int out_size, void* d_ws, size_t ws_size,
                              hipStream_t stream) {
    (void)in_sizes; (void)n_in; (void)out_size; (void)d_ws; (void)ws_size;
    const float* depth  = (const float*)d_in[0];
    const float* extrin = (const float*)d_in[1];
    const float* intrin = (const float*)d_in[2];
    const float* volume = (const float*)d_in[3];
    const float* wvol   = (const float*)d_in[4];
    const float* origin = (const float*)d_in[5];
    const float* resol  = (const float*)d_in[6];
    float* out = (float*)d_out;

    dim3 block(256);            // 8 wave32 waves
    dim3 grid(HW / 256);        // 300 blocks, exact cover -> EXEC all-ones for WMMA
    Extractor_56564719288936_kernel<<<grid, block, 0, stream>>>(
        depth, extrin, intrin, volume, wvol, origin, resol, out);
}